// SSM_14130442404221
// MI455X (gfx1250) — compile-verified
//
#include <hip/hip_runtime.h>

// SSM (Mamba-style selective scan), MI455X / gfx1250, wave32.
// Stage 1: fp32 WMMA projection GEMM (x @ Wx^T + bx, then dt GEMM) -> 96-float
//          record per (b,t) in workspace:  [B(16) | C(16) | dt(64)].
//          Wx staged zero-padded in LDS -> branch-free B operands.
// Stage 2: bandwidth-bound sequential scan, one h[d,n] per thread in registers,
//          coalesced hidden-state stores, 16-lane shuffle reduction for y.

#define BATCH   32
#define SEQ     4096
#define D_INNER 64
#define D_STATE 16
#define DT_RANK 4
#define NPROJ   (DT_RANK + 2 * D_STATE)   // 36
#define NPAD    48                        // 3 N-tiles of 16 (rows 36..47 zero)
#define REC     96                        // B(16) + C(16) + dt(64)
#define CLAMP_V 1000000.0f

typedef __attribute__((ext_vector_type(2))) float v2f;
typedef __attribute__((ext_vector_type(4))) float v4f;
typedef __attribute__((ext_vector_type(8))) float v8f;

// ---------------------------------------------------------------------------
// Stage 1: projection GEMMs via V_WMMA_F32_16X16X4_F32.
// Block = 256 threads = 8 waves; each wave handles one 16-row tile of
// X (BT x 64) against Wx^T (64 x 36, zero-padded to 48 cols = 3 N-tiles).
// ---------------------------------------------------------------------------
__global__ __launch_bounds__(256) void ssm_proj_kernel(
    const float* __restrict__ x,    // (BT, 64)
    const float* __restrict__ Wx,   // (36, 64)
    const float* __restrict__ bx,   // (36)
    const float* __restrict__ Wdt,  // (64, 4)
    const float* __restrict__ bdt,  // (64)
    float* __restrict__ rec)        // (BT, 96)
{
    __shared__ __attribute__((aligned(16))) float WxS[NPAD][D_INNER]; // 12 KB, zero-padded
    __shared__ __attribute__((aligned(16))) float dtin[8][16][4];     // xp cols 16..19 / wave

    const int tid  = threadIdx.x;
    const int w    = tid >> 5;        // wave in block
    const int lane = tid & 31;
    const int n    = lane & 15;       // N index (and M row for A loads)
    const int hh   = lane >> 4;       // half-wave select
    const long rb  = (long)blockIdx.x * 8 + w;   // 16-row tile index
    const long base = rb * 16;                   // first row of tile

    // Cooperative zero-padded copy of Wx into LDS (3072 floats / 256 threads).
    #pragma unroll
    for (int i = tid; i < NPAD * D_INNER; i += 256)
        WxS[i >> 6][i & 63] = (i < NPROJ * D_INNER) ? Wx[i] : 0.0f;
    __syncthreads();

    v8f c0 = {}; v8f c1 = {}; v8f c2 = {};

    const float* xrow = x + (base + n) * D_INNER;  // A: lane holds row m = lane&15

    #pragma unroll
    for (int k0 = 0; k0 < D_INNER; k0 += 4) {
        // A tile 16x4: vgpr0/1 hold K = 2*hh, 2*hh+1 for row n
        v2f a = *(const v2f*)(xrow + k0 + 2 * hh);
        // B tiles 4x16 from LDS (branch-free, one v2f per lane)
        v2f b0 = *(const v2f*)(&WxS[ 0 + n][k0 + 2 * hh]);
        v2f b1 = *(const v2f*)(&WxS[16 + n][k0 + 2 * hh]);
        v2f b2 = *(const v2f*)(&WxS[32 + n][k0 + 2 * hh]);
        c0 = __builtin_amdgcn_wmma_f32_16x16x4_f32(false, a, false, b0,
                                                   (short)0, c0, false, false);
        c1 = __builtin_amdgcn_wmma_f32_16x16x4_f32(false, a, false, b1,
                                                   (short)0, c1, false, false);
        c2 = __builtin_amdgcn_wmma_f32_16x16x4_f32(false, a, false, b2,
                                                   (short)0, c2, false, false);
    }

    // Scatter results. C/D layout: lane (hh*16+n) vgpr r holds element (M=r+8*hh, N=n).
    const float bx0 = bx[n];
    const float bx1 = bx[16 + n];
    const float bx2 = (32 + n < NPROJ) ? bx[32 + n] : 0.0f;
    #pragma unroll
    for (int r = 0; r < 8; ++r) {
        const int  m   = r + 8 * hh;
        const long row = base + m;
        // cols 0..15 -> B state input
        rec[row * REC + n] = c0[r] + bx0;
        // cols 16..19 -> dt-GEMM input (LDS); cols 20..31 -> C[0..11]
        const float v1 = c1[r] + bx1;
        if (n < 4) dtin[w][m][n] = v1;
        else       rec[row * REC + 16 + (n - 4)] = v1;
        // cols 32..35 -> C[12..15]
        if (n < 4)
            rec[row * REC + 16 + 12 + n] = c2[r] + bx2;
    }
    __syncthreads();

    // dt = xp[:,16:20] @ Wdt^T + bdt  (K=4: plain VALU, trivially cheap).
    // Each lane computes cols j1=lane, j2=lane+32 for all 16 rows.
    const int j1 = lane;
    const int j2 = lane + 32;
    const v4f w1 = *(const v4f*)(Wdt + j1 * 4);
    const v4f w2 = *(const v4f*)(Wdt + j2 * 4);
    const float b1 = bdt[j1], b2 = bdt[j2];
    #pragma unroll 4
    for (int m = 0; m < 16; ++m) {
        const v4f q = *(const v4f*)(&dtin[w][m][0]);   // broadcast LDS read
        const long row = base + m;
        rec[row * REC + 32 + j1] = b1 + q.x * w1.x + q.y * w1.y + q.z * w1.z + q.w * w1.w;
        rec[row * REC + 32 + j2] = b2 + q.x * w2.x + q.y * w2.y + q.z * w2.z + q.w * w2.w;
    }
}

// ---------------------------------------------------------------------------
// Stage 2: sequential scan. grid = 32 batches x 4 d-chunks; block = 256
// threads = 16 d x 16 n. One h[d,n] per thread; hidden store is a contiguous
// 1 KB per block per timestep (pure streaming -> HBM bound).
// ---------------------------------------------------------------------------
__global__ __launch_bounds__(256) void ssm_scan_kernel(
    const float* __restrict__ x,      // (B, SEQ, 64)
    const float* __restrict__ rec,    // (B*SEQ, 96)
    const float* __restrict__ A_log,  // (64, 16)
    const float* __restrict__ Dp,     // (64)
    float* __restrict__ out,          // (B, SEQ, 64)
    float* __restrict__ hid)          // (B, SEQ, 64, 16)
{
    const int tid = threadIdx.x;
    const int n   = tid & 15;
    const int dl  = tid >> 4;
    const int b   = blockIdx.x >> 2;
    const int dc  = blockIdx.x & 3;
    const int d   = dc * 16 + dl;

    const float A  = -expf(A_log[d * D_STATE + n]);
    const float Dd = Dp[d];

    const long  bt0  = (long)b * SEQ;
    const float* rp  = rec + bt0 * REC;
    const float* xp  = x + bt0 * D_INNER;
    float* hp = hid + bt0 * D_INNER * D_STATE;
    float* op = out + bt0 * D_INNER;

    float h = 0.0f;
    #pragma unroll 2
    for (int t = 0; t < SEQ; ++t) {
        const float* r = rp + (long)t * REC;
        const float Bn = r[n];
        const float Cn = r[16 + n];
        const float dt = r[32 + d];
        const float xi = xp[(long)t * D_INNER + d];

        const float dA = dt * A;
        const float dB = dt * Bn;
        h = h * dA + xi * dB;
        h = fminf(fmaxf(h, -CLAMP_V), CLAMP_V);

        hp[((long)t * D_INNER + d) * D_STATE + n] = h;   // coalesced stream

        float v = h * Cn;                                 // reduce over n (16 lanes)
        v += __shfl_xor(v, 1, 16);
        v += __shfl_xor(v, 2, 16);
        v += __shfl_xor(v, 4, 16);
        v += __shfl_xor(v, 8, 16);
        if (n == 0) op[(long)t * D_INNER + d] = v + Dd * xi;

        if (((t & 7) == 0) && (t + 16 < SEQ))             // stream-ahead hint
            __builtin_prefetch(rp + (long)(t + 16) * REC + (tid & 63), 0, 0);
    }
}

// ---------------------------------------------------------------------------
extern "C" void kernel_launch(void* const* d_in, const int* in_sizes, int n_in,
                              void* d_out, int out_size, void* d_ws, size_t ws_size,
                              hipStream_t stream) {
    (void)in_sizes; (void)n_in; (void)out_size; (void)ws_size;
    const float* x     = (const float*)d_in[0];
    const float* Wx    = (const float*)d_in[1];
    const float* bx    = (const float*)d_in[2];
    const float* Wdt   = (const float*)d_in[3];
    const float* bdt   = (const float*)d_in[4];
    const float* A_log = (const float*)d_in[5];
    const float* Dp    = (const float*)d_in[6];

    float* out = (float*)d_out;                                   // (32,4096,64)
    float* hid = out + (size_t)BATCH * SEQ * D_INNER;             // (32,4096,64,16)
    float* rec = (float*)d_ws;  // needs BATCH*SEQ*REC*4 = 48 MiB

    const int BT = BATCH * SEQ;                // 131072 rows
    dim3 g1(BT / 16 / 8);                      // 1024 blocks x 8 waves x 16-row tiles
    ssm_proj_kernel<<<g1, 256, 0, stream>>>(x, Wx, bx, Wdt, bdt, rec);

    dim3 g2(BATCH * 4);                        // 128 blocks
    ssm_scan_kernel<<<g2, 256, 0, stream>>>(x, rec, A_log, Dp, out, hid);
}